// TemporalEdgeMAE_69466801046053
// MI455X (gfx1250) — compile-verified
//
#include <hip/hip_runtime.h>

// Problem constants (from reference)
#define BB   4
#define CC   3
#define TT   16
#define VV   25
#define NN   400          // T*V
#define ENCD 256
#define DECD 128

typedef __attribute__((ext_vector_type(2))) float v2f;
typedef __attribute__((ext_vector_type(8))) float v8f;

// fp32 WMMA: D(16x16) = A(16x4) * B(4x16) + C   -> v_wmma_f32_16x16x4_f32
__device__ __forceinline__ v8f wmma_f32(v2f a, v2f b, v8f c) {
  return __builtin_amdgcn_wmma_f32_16x16x4_f32(
      /*neg_a=*/false, a, /*neg_b=*/false, b,
      /*c_mod=*/(short)0, c, /*reuse_a=*/false, /*reuse_b=*/false);
}

// ---------------------------------------------------------------------------
// 1) h[b,n,e] = sum_c x[b,c,n] * W_emb[c,e] + b_emb[e] + pos_emb[n,e]
// ---------------------------------------------------------------------------
__global__ __launch_bounds__(256) void k_embed(const float* __restrict__ x,
                                               const float* __restrict__ W_emb,
                                               const float* __restrict__ b_emb,
                                               const float* __restrict__ pos,
                                               float* __restrict__ h) {
  int idx = blockIdx.x * 256 + threadIdx.x;          // over B*N*ENC (exact)
  int e  = idx & (ENCD - 1);
  int bn = idx >> 8;                                  // ENC == 256
  int n  = bn % NN;
  int b  = bn / NN;
  float s = b_emb[e] + pos[(size_t)n * ENCD + e];
#pragma unroll
  for (int c = 0; c < CC; ++c)
    s += x[((size_t)b * CC + c) * (TT * VV) + n] * W_emb[(size_t)c * ENCD + e];
  h[idx] = s;
}

// ---------------------------------------------------------------------------
// 2) msg = (a * (1-mask)) @ h        M=400, K=400, Ncol=256
//    One 16x16 output tile per wave32; 100 fp32 WMMAs along K.
//    Masking is purely arithmetic (no EXEC divergence around WMMA).
// ---------------------------------------------------------------------------
__global__ __launch_bounds__(128) void k_adj(const float* __restrict__ a,
                                             const unsigned char* __restrict__ msk,
                                             const float* __restrict__ h,
                                             float* __restrict__ msg) {
  int wave = blockIdx.x * 4 + (threadIdx.x >> 5);     // 1600 waves total
  int lane = threadIdx.x & 31;
  int half = lane >> 4;                               // K half-split
  int l    = lane & 15;
  int b    = wave / 400;
  int rem  = wave % 400;                              // 25 x 16 tiles
  int tm   = rem >> 4, tn = rem & 15;
  int rowA = tm * 16 + l;                             // A-frag: lane = M row
  int colB = tn * 16 + l;                             // B-frag: lane = N col
  const float*         Ar = a   + ((size_t)b * NN + rowA) * NN;
  const unsigned char* Mr = msk + ((size_t)b * NN + rowA) * NN;
  const float*         Hb = h   + (size_t)b * NN * ENCD;
  v8f acc = {};
  for (int k0 = 0; k0 < NN; k0 += 4) {
    if ((k0 & 63) == 0)                               // uniform prefetch of A row
      __builtin_prefetch(Ar + k0 + 64, 0, 3);         // -> global_prefetch_b8
    int ka = k0 + half * 2;                           // even -> 8B aligned
    float2 a2 = *reinterpret_cast<const float2*>(Ar + ka);
    unsigned int m2 =
        (unsigned int)*reinterpret_cast<const unsigned short*>(Mr + ka);
    v2f av, bv;
    av.x = a2.x * (1.0f - (float)(m2 & 0xffu));       // branchless edge mask
    av.y = a2.y * (1.0f - (float)(m2 >> 8));
    bv.x = Hb[(size_t)ka * ENCD + colB];
    bv.y = Hb[(size_t)(ka + 1) * ENCD + colB];
    acc = wmma_f32(av, bv, acc);
  }
  float* Ob = msg + (size_t)b * NN * ENCD;
#pragma unroll
  for (int r = 0; r < 8; ++r)                         // D: M = r + 8*half
    Ob[(size_t)(tm * 16 + half * 8 + r) * ENCD + colB] = acc[r];
}

// ---------------------------------------------------------------------------
// 3) enc = relu(msg @ Wt1 + h @ Wt2)   M=400, K=256, Ncol=256 (dual GEMM)
// ---------------------------------------------------------------------------
__global__ __launch_bounds__(128) void k_enc(const float* __restrict__ msg,
                                             const float* __restrict__ h,
                                             const float* __restrict__ Wt1,
                                             const float* __restrict__ Wt2,
                                             float* __restrict__ enc) {
  int wave = blockIdx.x * 4 + (threadIdx.x >> 5);
  int lane = threadIdx.x & 31;
  int half = lane >> 4, l = lane & 15;
  int b   = wave / 400;
  int rem = wave % 400;
  int tm  = rem >> 4, tn = rem & 15;
  int rowA = tm * 16 + l;
  int colB = tn * 16 + l;
  const float* Mrow = msg + ((size_t)b * NN + rowA) * ENCD;
  const float* Hrow = h   + ((size_t)b * NN + rowA) * ENCD;
  v8f acc = {};
  for (int k0 = 0; k0 < ENCD; k0 += 4) {
    int ka = k0 + half * 2;
    float2 am = *reinterpret_cast<const float2*>(Mrow + ka);
    float2 ah = *reinterpret_cast<const float2*>(Hrow + ka);
    v2f a1, b1, a2, b2;
    a1.x = am.x;  a1.y = am.y;
    b1.x = Wt1[(size_t)ka * ENCD + colB];
    b1.y = Wt1[(size_t)(ka + 1) * ENCD + colB];
    acc = wmma_f32(a1, b1, acc);
    a2.x = ah.x;  a2.y = ah.y;
    b2.x = Wt2[(size_t)ka * ENCD + colB];
    b2.y = Wt2[(size_t)(ka + 1) * ENCD + colB];
    acc = wmma_f32(a2, b2, acc);
  }
  float* Ob = enc + (size_t)b * NN * ENCD;
#pragma unroll
  for (int r = 0; r < 8; ++r)
    Ob[(size_t)(tm * 16 + half * 8 + r) * ENCD + colB] = fmaxf(acc[r], 0.f);
}

// ---------------------------------------------------------------------------
// 4) dec = enc @ W_d + b_d      M=400, K=256, Ncol=128
// ---------------------------------------------------------------------------
__global__ __launch_bounds__(128) void k_dec(const float* __restrict__ enc,
                                             const float* __restrict__ W_d,
                                             const float* __restrict__ b_d,
                                             float* __restrict__ dec) {
  int wave = blockIdx.x * 4 + (threadIdx.x >> 5);     // 800 waves
  int lane = threadIdx.x & 31;
  int half = lane >> 4, l = lane & 15;
  int b   = wave / 200;
  int rem = wave % 200;                               // 25 x 8 tiles
  int tm  = rem >> 3, tn = rem & 7;
  int rowA = tm * 16 + l;
  int colB = tn * 16 + l;
  const float* Erow = enc + ((size_t)b * NN + rowA) * ENCD;
  v8f acc = {};
  for (int k0 = 0; k0 < ENCD; k0 += 4) {
    int ka = k0 + half * 2;
    float2 ae = *reinterpret_cast<const float2*>(Erow + ka);
    v2f av, bv;
    av.x = ae.x;  av.y = ae.y;
    bv.x = W_d[(size_t)ka * DECD + colB];
    bv.y = W_d[(size_t)(ka + 1) * DECD + colB];
    acc = wmma_f32(av, bv, acc);
  }
  float bias = b_d[colB];
  float* Ob = dec + (size_t)b * NN * DECD;
#pragma unroll
  for (int r = 0; r < 8; ++r)
    Ob[(size_t)(tm * 16 + half * 8 + r) * DECD + colB] = acc[r] + bias;
}

// ---------------------------------------------------------------------------
// 5) hi = dec @ W_e1[:DEC],  hj = dec @ W_e1[DEC:]   (shared A fragment)
// ---------------------------------------------------------------------------
__global__ __launch_bounds__(128) void k_hihj(const float* __restrict__ dec,
                                              const float* __restrict__ W_e1,
                                              float* __restrict__ hi,
                                              float* __restrict__ hj) {
  int wave = blockIdx.x * 4 + (threadIdx.x >> 5);     // 800 waves
  int lane = threadIdx.x & 31;
  int half = lane >> 4, l = lane & 15;
  int b   = wave / 200;
  int rem = wave % 200;
  int tm  = rem >> 3, tn = rem & 7;
  int rowA = tm * 16 + l;
  int colB = tn * 16 + l;
  const float* Drow = dec + ((size_t)b * NN + rowA) * DECD;
  const float* Wa   = W_e1;                           // rows 0..127   (Zi part)
  const float* Wb   = W_e1 + (size_t)DECD * DECD;     // rows 128..255 (Zj part)
  v8f accI = {}, accJ = {};
  for (int k0 = 0; k0 < DECD; k0 += 4) {
    int ka = k0 + half * 2;
    float2 ad = *reinterpret_cast<const float2*>(Drow + ka);
    v2f av, bi, bj;
    av.x = ad.x;  av.y = ad.y;
    bi.x = Wa[(size_t)ka * DECD + colB];
    bi.y = Wa[(size_t)(ka + 1) * DECD + colB];
    bj.x = Wb[(size_t)ka * DECD + colB];
    bj.y = Wb[(size_t)(ka + 1) * DECD + colB];
    accI = wmma_f32(av, bi, accI);
    accJ = wmma_f32(av, bj, accJ);
  }
  float* Oi = hi + (size_t)b * NN * DECD;
  float* Oj = hj + (size_t)b * NN * DECD;
#pragma unroll
  for (int r = 0; r < 8; ++r) {
    size_t o = (size_t)(tm * 16 + half * 8 + r) * DECD + colB;
    Oi[o] = accI[r];
    Oj[o] = accJ[r];
  }
}

// ---------------------------------------------------------------------------
// 6) edge_pred[b,i,j] = sum_d relu(hi[b,i,d]+hj[b,j,d]+b_e1[d]) * W_e2[d] + b_e2
//    + fused masked-MSE accumulation (branchless). One block per (b,i).
// ---------------------------------------------------------------------------
__global__ __launch_bounds__(128) void k_edge(const float* __restrict__ hi,
                                              const float* __restrict__ hj,
                                              const float* __restrict__ b_e1,
                                              const float* __restrict__ W_e2,
                                              const float* __restrict__ b_e2,
                                              const float* __restrict__ a,
                                              const unsigned char* __restrict__ msk,
                                              float* __restrict__ out,
                                              float* __restrict__ accum) {
  __shared__ float s_hib[DECD];
  __shared__ float s_w2[DECD];
  __shared__ float s_red[128];
  int bi  = blockIdx.x;                               // over B*N
  int b   = bi / NN;
  int i   = bi % NN;
  int tid = threadIdx.x;                              // 128 == DEC
  s_hib[tid] = hi[((size_t)b * NN + i) * DECD + tid] + b_e1[tid];
  s_w2[tid]  = W_e2[tid];
  __syncthreads();
  float be2 = b_e2[0];
  float sse = 0.f, cnt = 0.f;
  for (int j = tid; j < NN; j += 128) {
    const float* hjr = hj + ((size_t)b * NN + j) * DECD;
    float s = 0.f;
#pragma unroll 8
    for (int d = 0; d < DECD; d += 4) {
      float4 hv = *reinterpret_cast<const float4*>(hjr + d);
      s += fmaxf(s_hib[d]     + hv.x, 0.f) * s_w2[d];
      s += fmaxf(s_hib[d + 1] + hv.y, 0.f) * s_w2[d + 1];
      s += fmaxf(s_hib[d + 2] + hv.z, 0.f) * s_w2[d + 2];
      s += fmaxf(s_hib[d + 3] + hv.w, 0.f) * s_w2[d + 3];
    }
    float pred = s + be2;
    size_t o = ((size_t)b * NN + i) * NN + j;
    out[o] = pred;
    float m = (float)msk[o];                          // branchless masked MSE
    float diff = pred - a[o];
    sse += m * diff * diff;
    cnt += m;
  }
  // block reduction -> global atomics
  s_red[tid] = sse;
  __syncthreads();
  for (int off = 64; off > 0; off >>= 1) {
    if (tid < off) s_red[tid] += s_red[tid + off];
    __syncthreads();
  }
  if (tid == 0) atomicAdd(&accum[0], s_red[0]);
  __syncthreads();
  s_red[tid] = cnt;
  __syncthreads();
  for (int off = 64; off > 0; off >>= 1) {
    if (tid < off) s_red[tid] += s_red[tid + off];
    __syncthreads();
  }
  if (tid == 0) atomicAdd(&accum[1], s_red[0]);
}

__global__ void k_init(float* accum) { accum[0] = 0.f; accum[1] = 0.f; }
__global__ void k_fin(const float* accum, float* out) {
  out[(size_t)BB * NN * NN] = accum[0] / accum[1];
}

// ---------------------------------------------------------------------------
extern "C" void kernel_launch(void* const* d_in, const int* in_sizes, int n_in,
                              void* d_out, int out_size, void* d_ws, size_t ws_size,
                              hipStream_t stream) {
  const float*         x     = (const float*)d_in[0];
  const float*         a     = (const float*)d_in[1];
  const unsigned char* em    = (const unsigned char*)d_in[2];  // bool mask
  const float*         W_emb = (const float*)d_in[3];
  const float*         b_emb = (const float*)d_in[4];
  const float*         pos   = (const float*)d_in[5];
  const float*         Wt1   = (const float*)d_in[6];
  const float*         Wt2   = (const float*)d_in[7];
  const float*         W_d   = (const float*)d_in[8];
  const float*         b_d   = (const float*)d_in[9];
  const float*         W_e1  = (const float*)d_in[10];
  const float*         b_e1  = (const float*)d_in[11];
  const float*         W_e2  = (const float*)d_in[12];
  const float*         b_e2  = (const float*)d_in[13];
  float* out = (float*)d_out;

  float* ws  = (float*)d_ws;
  float* h   = ws;
  float* msg = h   + (size_t)BB * NN * ENCD;
  float* enc = msg + (size_t)BB * NN * ENCD;
  float* dec = enc + (size_t)BB * NN * ENCD;
  float* hi  = dec + (size_t)BB * NN * DECD;
  float* hj  = hi  + (size_t)BB * NN * DECD;
  float* acc = hj  + (size_t)BB * NN * DECD;

  k_embed<<<(BB * NN * ENCD) / 256, 256, 0, stream>>>(x, W_emb, b_emb, pos, h);
  k_adj  <<<400, 128, 0, stream>>>(a, em, h, msg);
  k_enc  <<<400, 128, 0, stream>>>(msg, h, Wt1, Wt2, enc);
  k_dec  <<<200, 128, 0, stream>>>(enc, W_d, b_d, dec);
  k_hihj <<<200, 128, 0, stream>>>(dec, W_e1, hi, hj);
  k_init <<<1, 1, 0, stream>>>(acc);
  k_edge <<<BB * NN, 128, 0, stream>>>(hi, hj, b_e1, W_e2, b_e2, a, em, out, acc);
  k_fin  <<<1, 1, 0, stream>>>(acc, out);
}